// DMol_architecture_21131239096354
// MI455X (gfx1250) — compile-verified
//
#include <hip/hip_runtime.h>
#include <hip/hip_bf16.h>

// ---------------------------------------------------------------------------
// DMol GNN forward for MI455X (gfx1250, wave32).
// Key rewrite: (x[src]) @ Wn == (x @ Wn)[src]  -> dense GEMM + cheap gather.
// All dense GEMMs use V_WMMA_F32_16X16X4_F32 (full fp32 precision).
// Edge aggregation: per-graph 64x256 fp32 tile in LDS, ds_add_f32 atomics.
// ---------------------------------------------------------------------------

typedef __attribute__((ext_vector_type(2))) float v2f;
typedef __attribute__((ext_vector_type(8))) float v8f;

#define DM_B 256        // graphs
#define DM_N 64         // nodes / graph
#define DM_D 256        // hidden dim
#define DM_T (DM_B * DM_N)
#define DM_E 524288
#define DM_EPG (DM_E / DM_B)   // 2048 edges per graph
#define RRELU_SLOPE 0.2291666666666667f

static __device__ __forceinline__ v8f wmma_f32(v2f a, v2f b, v8f c) {
    return __builtin_amdgcn_wmma_f32_16x16x4_f32(false, a, false, b, (short)0, c,
                                                 false, false);
}

// ---------------------------------------------------------------------------
// Generic fp32 WMMA GEMM:  C = act(A[M,K] @ W[K,N] + bias + add)
// Block tile 64(M) x 128(N), BK=16, 8 waves of 32x32.
// Requires M%64==0, N%128==0; K arbitrary (zero-padded in LDS).
// act: 0=none, 1=relu, 2=rrelu(eval slope)
// ---------------------------------------------------------------------------
__global__ __launch_bounds__(256)
void gemm_wmma_kernel(const float* __restrict__ A, const float* __restrict__ W,
                      const float* __restrict__ bias, const float* __restrict__ add,
                      float* __restrict__ C, int M, int N, int K, int act) {
    __shared__ float sA[64][16];
    __shared__ float sB[16][128];

    const int m0   = blockIdx.x * 64;
    const int n0   = blockIdx.y * 128;
    const int tid  = threadIdx.x;
    const int wave = tid >> 5;
    const int lane = tid & 31;
    const int hl   = lane >> 4;     // half-wave select
    const int l16  = lane & 15;
    const int wm   = (wave >> 2) * 32;   // wave M offset in block tile
    const int wn   = (wave & 3) * 32;    // wave N offset in block tile

    v8f acc[2][2];
#pragma unroll
    for (int i = 0; i < 2; ++i)
#pragma unroll
        for (int j = 0; j < 2; ++j) acc[i][j] = (v8f){};

    for (int k0 = 0; k0 < K; k0 += 16) {
        // stage A tile 64x16
#pragma unroll
        for (int i = 0; i < 4; ++i) {
            int idx = tid + i * 256;
            int r = idx >> 4, c = idx & 15;
            sA[r][c] = (k0 + c < K) ? A[(size_t)(m0 + r) * K + k0 + c] : 0.f;
        }
        // stage B tile 16x128
#pragma unroll
        for (int i = 0; i < 8; ++i) {
            int idx = tid + i * 256;
            int r = idx >> 7, c = idx & 127;
            sB[r][c] = (k0 + r < K) ? W[(size_t)(k0 + r) * N + n0 + c] : 0.f;
        }
        __syncthreads();
#pragma unroll
        for (int kk = 0; kk < 16; kk += 4) {
            const int ka = kk + 2 * hl;
            v2f a0, a1, b0, b1;
            a0.x = sA[wm + l16][ka];       a0.y = sA[wm + l16][ka + 1];
            a1.x = sA[wm + 16 + l16][ka];  a1.y = sA[wm + 16 + l16][ka + 1];
            b0.x = sB[ka][wn + l16];       b0.y = sB[ka + 1][wn + l16];
            b1.x = sB[ka][wn + 16 + l16];  b1.y = sB[ka + 1][wn + 16 + l16];
            acc[0][0] = wmma_f32(a0, b0, acc[0][0]);
            acc[0][1] = wmma_f32(a0, b1, acc[0][1]);
            acc[1][0] = wmma_f32(a1, b0, acc[1][0]);
            acc[1][1] = wmma_f32(a1, b1, acc[1][1]);
        }
        __syncthreads();
    }

#pragma unroll
    for (int mi = 0; mi < 2; ++mi)
#pragma unroll
        for (int ni = 0; ni < 2; ++ni) {
            const int col = n0 + wn + ni * 16 + l16;
            const float bcol = bias ? bias[col] : 0.f;
#pragma unroll
            for (int r = 0; r < 8; ++r) {
                const int row = m0 + wm + mi * 16 + hl * 8 + r;
                float v = acc[mi][ni][r] + bcol;
                if (add) v += add[(size_t)row * N + col];
                if (act == 1)      v = v > 0.f ? v : 0.f;
                else if (act == 2) v = v >= 0.f ? v : RRELU_SLOPE * v;
                C[(size_t)row * N + col] = v;
            }
        }
}

// ---------------------------------------------------------------------------
// Edge conv: one workgroup per graph.
//   msg_e = relu(hn[src[e]] + ea[e] @ We + bm)   (ea@We via WMMA, K=16)
//   agg[dst[e]] += msg_e   (ds_add_f32 into 64x256 LDS tile)
// ---------------------------------------------------------------------------
__global__ __launch_bounds__(256)
void edge_conv_kernel(const float* __restrict__ hn, const float* __restrict__ ea,
                      const int* __restrict__ src, const int* __restrict__ dst,
                      const float* __restrict__ We, const float* __restrict__ bm,
                      float* __restrict__ agg) {
    __shared__ float sAgg[DM_N][DM_D];   // exactly 64 KB

    const int g    = blockIdx.x;
    const int tid  = threadIdx.x;
    const int wave = tid >> 5;
    const int lane = tid & 31;
    const int hl   = lane >> 4;
    const int l16  = lane & 15;
    const int edge0 = g * DM_EPG;
    const int node0 = g * DM_N;

    for (int i = tid; i < DM_N * DM_D; i += 256) ((float*)sAgg)[i] = 0.f;
    __syncthreads();

    for (int nt = 0; nt < 16; ++nt) {           // 16 column tiles of 16
        const int n = nt * 16 + l16;
        // hoist We fragments for K=16 (4 wmma k-steps) and bias
        v2f bf[4];
#pragma unroll
        for (int kk = 0; kk < 4; ++kk) {
            const int kr = kk * 4 + 2 * hl;
            bf[kk].x = We[kr * DM_D + n];
            bf[kk].y = We[(kr + 1) * DM_D + n];
        }
        const float bias = bm[n];

        for (int it = 0; it < 16; ++it) {       // 16 groups of 16 edges / wave
            const int ebase = edge0 + it * 128 + wave * 16;
            const float* eap = ea + (size_t)(ebase + l16) * 16;
            v2f af[4];
#pragma unroll
            for (int kk = 0; kk < 4; ++kk) {
                const int kc = kk * 4 + 2 * hl;
                af[kk].x = eap[kc];
                af[kk].y = eap[kc + 1];
            }
            v8f c = (v8f){};
#pragma unroll
            for (int kk = 0; kk < 4; ++kk) c = wmma_f32(af[kk], bf[kk], c);

#pragma unroll
            for (int r = 0; r < 8; ++r) {
                const int m = r + hl * 8;
                const int e = ebase + m;
                const int s = src[e];
                const int d = dst[e] - node0;
                float v = c[r] + bias + hn[(size_t)s * DM_D + n];
                v = v > 0.f ? v : 0.f;
                atomicAdd(&sAgg[d][n], v);
            }
        }
    }
    __syncthreads();
    float4* dstp = (float4*)agg + (size_t)g * (DM_N * DM_D / 4);
    const float4* srcp = (const float4*)sAgg;
    for (int i = tid; i < DM_N * DM_D / 4; i += 256) dstp[i] = srcp[i];
}

// ---------------------------------------------------------------------------
// dot_pool: per graph S = X1 @ X2^T (64x64), emit max and mean.
// ---------------------------------------------------------------------------
__global__ __launch_bounds__(256)
void dot_pool_kernel(const float* __restrict__ x1, const float* __restrict__ x2,
                     float* __restrict__ fusion, int step) {
    __shared__ float smax[8], ssum[8];
    const int g    = blockIdx.x;
    const int tid  = threadIdx.x;
    const int wave = tid >> 5;
    const int lane = tid & 31;
    const int hl   = lane >> 4;
    const int l16  = lane & 15;
    const float* X1 = x1 + (size_t)g * DM_N * DM_D;
    const float* X2 = x2 + (size_t)g * DM_N * DM_D;

    float wmax = -3.402823466e38f, wsum = 0.f;
#pragma unroll
    for (int tt = 0; tt < 2; ++tt) {           // 8 waves x 2 = 16 tiles
        const int t = wave * 2 + tt;
        const int mi = t >> 2, ni = t & 3;
        const float* ar = X1 + (size_t)(mi * 16 + l16) * DM_D;
        const float* br = X2 + (size_t)(ni * 16 + l16) * DM_D;  // B = X2^T
        v8f c = (v8f){};
        for (int k0 = 0; k0 < DM_D; k0 += 4) {
            const int ka = k0 + 2 * hl;
            v2f a, b;
            a.x = ar[ka]; a.y = ar[ka + 1];
            b.x = br[ka]; b.y = br[ka + 1];
            c = wmma_f32(a, b, c);
        }
#pragma unroll
        for (int r = 0; r < 8; ++r) { float v = c[r]; wmax = fmaxf(wmax, v); wsum += v; }
    }
#pragma unroll
    for (int off = 16; off > 0; off >>= 1) {   // wave32 reduction
        wmax = fmaxf(wmax, __shfl_xor(wmax, off, 32));
        wsum += __shfl_xor(wsum, off, 32);
    }
    if (lane == 0) { smax[wave] = wmax; ssum[wave] = wsum; }
    __syncthreads();
    if (tid == 0) {
        float m = smax[0], s = ssum[0];
        for (int w = 1; w < 8; ++w) { m = fmaxf(m, smax[w]); s += ssum[w]; }
        fusion[g * 6 + step * 2]     = m;
        fusion[g * 6 + step * 2 + 1] = s * (1.f / (DM_N * DM_N));
    }
}

// ---------------------------------------------------------------------------
// readout: [mean(256) | sum(256) | top3-rows-by-last-col (3*256)] per graph
// ---------------------------------------------------------------------------
__global__ __launch_bounds__(256)
void readout_kernel(const float* __restrict__ x, float* __restrict__ ro) {
    __shared__ int topidx[3];
    const int g = blockIdx.x, tid = threadIdx.x;
    const float* X = x + (size_t)g * DM_N * DM_D;
    if (tid == 0) {
        int i0 = 0, i1 = 0, i2 = 0;
        float v0 = -3.402823466e38f, v1 = v0, v2 = v0;
        for (int n = 0; n < DM_N; ++n) {
            float v = X[(size_t)n * DM_D + (DM_D - 1)];
            if (v > v0)      { v2 = v1; i2 = i1; v1 = v0; i1 = i0; v0 = v; i0 = n; }
            else if (v > v1) { v2 = v1; i2 = i1; v1 = v;  i1 = n; }
            else if (v > v2) { v2 = v;  i2 = n; }
        }
        topidx[0] = i0; topidx[1] = i1; topidx[2] = i2;
    }
    __syncthreads();
    float s = 0.f;
    for (int n = 0; n < DM_N; ++n) s += X[(size_t)n * DM_D + tid];
    float* R = ro + (size_t)g * (5 * DM_D);
    R[tid]           = s * (1.f / DM_N);
    R[DM_D + tid]    = s;
#pragma unroll
    for (int j = 0; j < 3; ++j)
        R[2 * DM_D + j * DM_D + tid] = X[(size_t)topidx[j] * DM_D + tid];
}

// out0 = [outm1(256) | outm2(256) | fusion(6)]  -> [B, 518]
__global__ void concat_kernel(const float* __restrict__ om1, const float* __restrict__ om2,
                              const float* __restrict__ fus, float* __restrict__ cat) {
    const int i = blockIdx.x * blockDim.x + threadIdx.x;
    if (i >= DM_B * 518) return;
    const int b = i / 518, c = i % 518;
    float v;
    if (c < 256)      v = om1[b * 256 + c];
    else if (c < 512) v = om2[b * 256 + (c - 256)];
    else              v = fus[b * 6 + (c - 512)];
    cat[i] = v;
}

// final: out[b, j] = h[b,:] @ Wo2[:, j] + bo2[j]   (N=2, VALU)
__global__ void final_kernel(const float* __restrict__ h, const float* __restrict__ Wo2,
                             const float* __restrict__ bo2, float* __restrict__ out) {
    const int i = blockIdx.x * blockDim.x + threadIdx.x;
    if (i >= DM_B * 2) return;
    const int b = i >> 1, j = i & 1;
    float s = bo2[j];
    for (int d = 0; d < DM_D; ++d) s += h[(size_t)b * DM_D + d] * Wo2[d * 2 + j];
    out[i] = s;
}

// ---------------------------------------------------------------------------
extern "C" void kernel_launch(void* const* d_in, const int* in_sizes, int n_in,
                              void* d_out, int out_size, void* d_ws, size_t ws_size,
                              hipStream_t stream) {
    const float* x1   = (const float*)d_in[0];
    const int*   ei1  = (const int*)d_in[1];
    const float* ea1  = (const float*)d_in[2];
    const float* x2   = (const float*)d_in[4];
    const int*   ei2  = (const int*)d_in[5];
    const float* ea2  = (const float*)d_in[6];
    const float* W0_1 = (const float*)d_in[8];
    const float* b0_1 = (const float*)d_in[9];
    const float* W0_2 = (const float*)d_in[10];
    const float* b0_2 = (const float*)d_in[11];
    const float* Wn1  = (const float*)d_in[12];
    const float* We1  = (const float*)d_in[13];
    const float* bm1  = (const float*)d_in[14];
    const float* Ws1  = (const float*)d_in[15];
    const float* bs1  = (const float*)d_in[16];
    const float* Wn2  = (const float*)d_in[17];
    const float* We2  = (const float*)d_in[18];
    const float* bm2  = (const float*)d_in[19];
    const float* Ws2  = (const float*)d_in[20];
    const float* bs2  = (const float*)d_in[21];
    const float* Wf1  = (const float*)d_in[22];
    const float* bf1  = (const float*)d_in[23];
    const float* Wf2  = (const float*)d_in[24];
    const float* bf2  = (const float*)d_in[25];
    const float* Wo1  = (const float*)d_in[26];
    const float* bo1  = (const float*)d_in[27];
    const float* Wo2  = (const float*)d_in[28];
    const float* bo2  = (const float*)d_in[29];

    // workspace carve-up (floats, 16B aligned)
    float* ws = (float*)d_ws;
    size_t off = 0;
    auto carve = [&](size_t n) { float* p = ws + off; off += (n + 3) & ~(size_t)3; return p; };
    float* xm1a = carve((size_t)DM_T * DM_D);
    float* xm1b = carve((size_t)DM_T * DM_D);
    float* xm2a = carve((size_t)DM_T * DM_D);
    float* xm2b = carve((size_t)DM_T * DM_D);
    float* hn   = carve((size_t)DM_T * DM_D);
    float* agg  = carve((size_t)DM_T * DM_D);
    float* ro1  = carve((size_t)DM_B * 5 * DM_D);
    float* ro2  = carve((size_t)DM_B * 5 * DM_D);
    float* om1  = carve((size_t)DM_B * DM_D);
    float* om2  = carve((size_t)DM_B * DM_D);
    float* fus  = carve((size_t)DM_B * 6);
    float* cat  = carve((size_t)DM_B * 518);
    float* o1   = carve((size_t)DM_B * DM_D);

    auto gemm = [&](const float* A, const float* W, const float* bias,
                    const float* add, float* C, int M, int N, int K, int act) {
        dim3 grid(M / 64, N / 128);
        gemm_wmma_kernel<<<grid, 256, 0, stream>>>(A, W, bias, add, C, M, N, K, act);
    };

    // initial embeddings: xm = rrelu(x @ W0 + b0)
    gemm(x1, W0_1, b0_1, nullptr, xm1a, DM_T, DM_D, 128, 2);
    gemm(x2, W0_2, b0_2, nullptr, xm2a, DM_T, DM_D, 128, 2);

    const int* s1 = ei1;           const int* d1 = ei1 + DM_E;
    const int* s2 = ei2;           const int* d2 = ei2 + DM_E;

    float* c1 = xm1a; float* n1 = xm1b;
    float* c2 = xm2a; float* n2 = xm2b;
    for (int step = 0; step < 3; ++step) {
        // molecule 1
        gemm(c1, Wn1, nullptr, nullptr, hn, DM_T, DM_D, DM_D, 0);
        edge_conv_kernel<<<DM_B, 256, 0, stream>>>(hn, ea1, s1, d1, We1, bm1, agg);
        gemm(c1, Ws1, bs1, agg, n1, DM_T, DM_D, DM_D, 1);
        // molecule 2
        gemm(c2, Wn2, nullptr, nullptr, hn, DM_T, DM_D, DM_D, 0);
        edge_conv_kernel<<<DM_B, 256, 0, stream>>>(hn, ea2, s2, d2, We2, bm2, agg);
        gemm(c2, Ws2, bs2, agg, n2, DM_T, DM_D, DM_D, 1);
        // fusion features on the updated states
        dot_pool_kernel<<<DM_B, 256, 0, stream>>>(n1, n2, fus, step);
        float* t;
        t = c1; c1 = n1; n1 = t;
        t = c2; c2 = n2; n2 = t;
    }

    readout_kernel<<<DM_B, 256, 0, stream>>>(c1, ro1);
    readout_kernel<<<DM_B, 256, 0, stream>>>(c2, ro2);
    gemm(ro1, Wf1, bf1, nullptr, om1, DM_B, DM_D, 5 * DM_D, 0);
    gemm(ro2, Wf2, bf2, nullptr, om2, DM_B, DM_D, 5 * DM_D, 0);

    concat_kernel<<<(DM_B * 518 + 255) / 256, 256, 0, stream>>>(om1, om2, fus, cat);
    gemm(cat, Wo1, bo1, nullptr, o1, DM_B, DM_D, 518, 0);
    final_kernel<<<2, 256, 0, stream>>>(o1, Wo2, bo2, (float*)d_out);
}